// L1LossSMPLCamVMRLE_90245852824002
// MI455X (gfx1250) — compile-verified
//
#include <hip/hip_runtime.h>
#include <hip/hip_bf16.h>

typedef __attribute__((ext_vector_type(2))) float v2f;
typedef __attribute__((ext_vector_type(8))) float v8f;

#define WAVES_PER_BLOCK   8
#define BLOCK_THREADS     256
#define SAMPLES_PER_WAVE  16
#define SAMPLES_PER_BLOCK (WAVES_PER_BLOCK * SAMPLES_PER_WAVE)

enum {
  ACC_BETA = 0, ACC_THETA, ACC_TWIST, ACC_XYZ_S, ACC_XYZ_W,
  ACC_UVD_W, ACC_UVD_TOT, ACC_UVD_MASK, ACC_UVD_USED,
  ACC_TRANS, ACC_SCALE, N_ACC
};

__global__ void init_acc_kernel(float* acc) {
  if (threadIdx.x < N_ACC) acc[threadIdx.x] = 0.0f;
}

__device__ __forceinline__ float wave_reduce_add(float v) {
  #pragma unroll
  for (int off = 16; off > 0; off >>= 1) v += __shfl_down(v, off, 32);
  return v;
}

__global__ __launch_bounds__(BLOCK_THREADS)
void smpl_loss_main_kernel(
    const float* __restrict__ pred_shape,      // B,10
    const float* __restrict__ pred_theta_mats, // B,216
    const float* __restrict__ pred_phi,        // B,46
    const float* __restrict__ pred_xyz_jts,    // B,87
    const float* __restrict__ nf_loss,         // B,288
    const float* __restrict__ sigma,           // B,288
    const float* __restrict__ pred_uvd_jts,    // B,87
    const float* __restrict__ pred_uvd_mks,    // B,201
    const float* __restrict__ cam_trans,       // B,2
    const float* __restrict__ cam_scale,       // B,1
    const float* __restrict__ tgt_smpl_w,      // B,1
    const float* __restrict__ tgt_beta,        // B,10
    const float* __restrict__ tgt_theta,       // B,216
    const float* __restrict__ tgt_theta_w,     // B,216
    const float* __restrict__ tgt_twist,       // B,46
    const float* __restrict__ tgt_twist_w,     // B,46
    const float* __restrict__ tgt_xyz24,       // B,72
    const float* __restrict__ tgt_xyz_w24,     // B,72
    const float* __restrict__ tgt_uvd29,       // B,87
    const float* __restrict__ tgt_uvd67,       // B,201
    const float* __restrict__ tgt_w29,         // B,87
    const float* __restrict__ tgt_w67,         // B,201
    const float* __restrict__ camera_trans,    // B,2
    const float* __restrict__ camera_scale,    // B,1
    int Bn,
    float* __restrict__ acc)
{
  __shared__ float sred[WAVES_PER_BLOCK][N_ACC];

  const int tid  = threadIdx.x;
  const int wave = tid >> 5;
  const int lane = tid & 31;
  const int half = lane >> 4;   // 0 -> K={0,1}, 1 -> K={2,3}  (A 16x4 f32 layout)
  const int mrow = lane & 15;   // A-matrix row (sample within 16-sample tile)
  const int s0   = (blockIdx.x * WAVES_PER_BLOCK + wave) * SAMPLES_PER_WAVE;
  const int s    = s0 + mrow;
  const bool wave_active = (s0 + SAMPLES_PER_WAVE) <= Bn;

  float pBeta = 0.f, pTheta = 0.f, pTwist = 0.f, pXyzS = 0.f, pXyzW = 0.f, pUvdW = 0.f;
  float pTot = 0.f, pMask = 0.f, pUsed = 0.f, pTrans = 0.f, pScale = 0.f;

  const float logAMP = -0.9189385332046727f;  // log(1/sqrt(2*pi))
  const float SQRT2  = 1.4142135623730951f;
  const v2f bones = {1.0f, 1.0f};             // B = ones(4x16): D[m][n] = sum_k A[m][k] + C

  if (wave_active) {
    // per-lane row base pointers (hoisted out of the loops)
    const float* nf_row = nf_loss       + (size_t)s * 288;
    const float* sg_row = sigma         + (size_t)s * 288;
    const float* pj_row = pred_uvd_jts  + (size_t)s * 87;
    const float* tj_row = tgt_uvd29     + (size_t)s * 87;
    const float* wj_row = tgt_w29       + (size_t)s * 87;
    const float* pm_row = pred_uvd_mks  + (size_t)s * 201 - 87;  // index by e directly
    const float* tm_row = tgt_uvd67     + (size_t)s * 201 - 87;
    const float* wm_row = tgt_w67       + (size_t)s * 201 - 87;

    // ---------- Phase A: loss_uvd elementwise + per-sample sums via WMMA ----------
    v8f cuvd = {0.f, 0.f, 0.f, 0.f, 0.f, 0.f, 0.f, 0.f};

    // branchless elementwise evaluator for a contiguous pair (e0 even -> nf/sg 8B aligned)
    auto uvd_pair = [&](int e0, const float* pr, const float* tr, const float* wr) -> v2f {
      const v2f nfv = *(const v2f*)(nf_row + e0);
      const v2f sgv = *(const v2f*)(sg_row + e0);
      v2f a;
      #pragma unroll
      for (int j = 0; j < 2; ++j) {
        const float p = pr[e0 + j];
        const float t = tr[e0 + j];
        const float w = wr[e0 + j];
        const float q = (logf(sgv[j]) - logAMP +
                         fabsf(t - p) / (SQRT2 * sgv[j] + 1e-9f)) * w;
        pUvdW += w;
        a[j] = nfv[j] * w + q;
      }
      return a;
    };

    for (int it = 0; it < 21; ++it) {          // e in [0,84): all in 87-wide arrays
      const int e0 = it * 4 + half * 2;
      const v2f a = uvd_pair(e0, pj_row, tj_row, wj_row);
      cuvd = __builtin_amdgcn_wmma_f32_16x16x4_f32(
          false, a, false, bones, (short)0, cuvd, false, false);
    }
    { // it = 21: e in {84,85,86,87}; only e==87 (half==1, j==1) is in the 201-wide arrays
      const int e0 = 84 + half * 2;
      const v2f nfv = *(const v2f*)(nf_row + e0);
      const v2f sgv = *(const v2f*)(sg_row + e0);
      v2f a;
      #pragma unroll
      for (int j = 0; j < 2; ++j) {
        const bool in_j = (j == 0) | (half == 0);          // lane-uniform cndmask select
        const float* pr = in_j ? pj_row : pm_row;
        const float* tr = in_j ? tj_row : tm_row;
        const float* wr = in_j ? wj_row : wm_row;
        const float p = pr[e0 + j];
        const float t = tr[e0 + j];
        const float w = wr[e0 + j];
        const float q = (logf(sgv[j]) - logAMP +
                         fabsf(t - p) / (SQRT2 * sgv[j] + 1e-9f)) * w;
        pUvdW += w;
        a[j] = nfv[j] * w + q;
      }
      cuvd = __builtin_amdgcn_wmma_f32_16x16x4_f32(
          false, a, false, bones, (short)0, cuvd, false, false);
    }
    for (int it = 22; it < 72; ++it) {         // e in [88,288): all in 201-wide arrays
      const int e0 = it * 4 + half * 2;
      const v2f a = uvd_pair(e0, pm_row, tm_row, wm_row);
      cuvd = __builtin_amdgcn_wmma_f32_16x16x4_f32(
          false, a, false, bones, (short)0, cuvd, false, false);
    }

    // D column 0: lane 0 holds rows M=0..7 in v0..v7, lane 16 holds M=8..15
    if (lane == 0 || lane == 16) {
      #pragma unroll
      for (int r = 0; r < 8; ++r) {
        const float ps = cuvd[r];
        pTot += ps;
        if (ps <= 0.f) { pMask += ps; pUsed += 1.f; }
      }
    }

    // ---------- Phase B: xyz weighted-L1 + per-sample txw sums (for sw2) ----------
    const float* xw_row = tgt_xyz_w24  + (size_t)s * 72;
    const float* tx_row = tgt_xyz24    + (size_t)s * 72;
    const float* px_row = pred_xyz_jts + (size_t)s * 87;   // first 72 of 87

    v8f cw = {0.f, 0.f, 0.f, 0.f, 0.f, 0.f, 0.f, 0.f};
    for (int it = 0; it < 18; ++it) {          // 18 * 4 = 72 xyz elements/sample
      const int e0 = it * 4 + half * 2;
      const v2f wv = *(const v2f*)(xw_row + e0);   // stride 72, e0 even -> aligned
      const v2f tv = *(const v2f*)(tx_row + e0);
      v2f a;
      #pragma unroll
      for (int j = 0; j < 2; ++j) {
        const float px = px_row[e0 + j];
        pXyzS += 64.0f * fabsf(px - tv[j]) * wv[j];
        pXyzW += wv[j];
        a[j] = wv[j];
      }
      cw = __builtin_amdgcn_wmma_f32_16x16x4_f32(
          false, a, false, bones, (short)0, cw, false, false);
    }
    if (lane == 0 || lane == 16) {
      const int rbase = (lane >> 4) * 8;
      #pragma unroll
      for (int r = 0; r < 8; ++r) {
        const int   sr  = s0 + rbase + r;
        const float sw2 = (cw[r] > 3.0f) ? 1.0f : 0.0f;
        const float d0 = (cam_trans[sr * 2 + 0] - camera_trans[sr * 2 + 0]) * sw2;
        const float d1 = (cam_trans[sr * 2 + 1] - camera_trans[sr * 2 + 1]) * sw2;
        pTrans += d0 * d0 + d1 * d1;
        const float ds = (cam_scale[sr] - camera_scale[sr]) * sw2;
        pScale += ds * ds;
      }
    }
  }

  // ---------- Phase C: flat, coalesced MSE partials over this block's samples ----------
  const int sBase = blockIdx.x * SAMPLES_PER_BLOCK;

  for (int i = tid; i < SAMPLES_PER_BLOCK * 10; i += BLOCK_THREADS) {   // beta
    const int smp = sBase + i / 10;
    if (smp >= Bn) break;
    const int k = i % 10;
    const float sw = tgt_smpl_w[smp];
    const float d = (pred_shape[smp * 10 + k] - tgt_beta[smp * 10 + k]) * sw;
    pBeta += d * d;
  }
  for (int i = tid; i < SAMPLES_PER_BLOCK * 216; i += BLOCK_THREADS) {  // theta
    const int smp = sBase + i / 216;
    if (smp >= Bn) break;
    const int k = i % 216;
    const float sw = tgt_smpl_w[smp];
    const float m = sw * tgt_theta_w[smp * 216 + k];
    const float d = (pred_theta_mats[smp * 216 + k] - tgt_theta[smp * 216 + k]) * m;
    pTheta += d * d;
  }
  for (int i = tid; i < SAMPLES_PER_BLOCK * 46; i += BLOCK_THREADS) {   // twist
    const int smp = sBase + i / 46;
    if (smp >= Bn) break;
    const int k = i % 46;
    const float w = tgt_twist_w[smp * 46 + k];
    const float d = (pred_phi[smp * 46 + k] - tgt_twist[smp * 46 + k]) * w;
    pTwist += d * d;
  }

  // ---------- Wave shuffle reduction -> LDS -> one atomic per accumulator ----------
  float vals[N_ACC] = { pBeta, pTheta, pTwist, pXyzS, pXyzW,
                        pUvdW, pTot, pMask, pUsed, pTrans, pScale };
  #pragma unroll
  for (int ai = 0; ai < N_ACC; ++ai) {
    const float r = wave_reduce_add(vals[ai]);
    if (lane == 0) sred[wave][ai] = r;
  }
  __syncthreads();
  if (tid < N_ACC) {
    float r = 0.f;
    #pragma unroll
    for (int wv = 0; wv < WAVES_PER_BLOCK; ++wv) r += sred[wv][tid];
    atomicAdd(&acc[tid], r);
  }
}

__global__ void finalize_kernel(const float* __restrict__ acc,
                                const int* __restrict__ epoch_ptr,
                                int Bn, float* __restrict__ out) {
  if (threadIdx.x != 0 || blockIdx.x != 0) return;
  const int   epoch = epoch_ptr[0];
  const float Bf = (float)Bn;

  const float loss_beta  = acc[ACC_BETA]  / (Bf * 10.0f);
  const float loss_theta = acc[ACC_THETA] / (Bf * 216.0f);
  const float loss_twist = acc[ACC_TWIST] / (Bf * 46.0f);

  const float ws = acc[ACC_XYZ_W];
  const float loss_xyz = (ws > 0.0f) ? acc[ACC_XYZ_S] / ws : acc[ACC_XYZ_S];

  float used, masked;
  if (epoch > 0) { used = acc[ACC_UVD_USED]; masked = acc[ACC_UVD_MASK]; }
  else           { used = Bf;                masked = acc[ACC_UVD_TOT];  }
  const float loss_uvd = (acc[ACC_UVD_W] > 0.0f) ? (masked / used) : acc[ACC_UVD_TOT];

  const float trans_loss = acc[ACC_TRANS] / (Bf * 2.0f);
  const float scale_loss = acc[ACC_SCALE] / Bf;

  float loss = loss_beta * 1.0f + loss_theta * 0.01f + loss_twist * 0.01f;
  if (epoch > 2) loss += loss_xyz * 1.0f;   // XYZ24_W, PRETRAIN_EPOCH = 2
  loss += loss_uvd * 1.0f;                  // UVD24_W
  const float cam_coef = (epoch > 2) ? 0.1f : 1.0f;
  loss += cam_coef * (trans_loss + scale_loss);

  out[0] = loss;
  out[1] = loss_uvd;
  out[2] = loss_beta;
  out[3] = loss_theta;
  out[4] = loss_twist;
  out[5] = trans_loss;
  out[6] = scale_loss;
}

extern "C" void kernel_launch(void* const* d_in, const int* in_sizes, int n_in,
                              void* d_out, int out_size, void* d_ws, size_t ws_size,
                              hipStream_t stream) {
  const int Bn = in_sizes[0] / 10;   // pred_shape is (B, 10)
  float* acc = (float*)d_ws;

  init_acc_kernel<<<1, 32, 0, stream>>>(acc);

  const int nBlocks = (Bn + SAMPLES_PER_BLOCK - 1) / SAMPLES_PER_BLOCK;
  smpl_loss_main_kernel<<<nBlocks, BLOCK_THREADS, 0, stream>>>(
      (const float*)d_in[0],  (const float*)d_in[1],  (const float*)d_in[2],
      (const float*)d_in[3],  (const float*)d_in[4],  (const float*)d_in[5],
      (const float*)d_in[6],  (const float*)d_in[7],  (const float*)d_in[8],
      (const float*)d_in[9],  (const float*)d_in[10], (const float*)d_in[11],
      (const float*)d_in[12], (const float*)d_in[13], (const float*)d_in[14],
      (const float*)d_in[15], (const float*)d_in[16], (const float*)d_in[17],
      (const float*)d_in[18], (const float*)d_in[19], (const float*)d_in[20],
      (const float*)d_in[21], (const float*)d_in[22], (const float*)d_in[23],
      Bn, acc);

  finalize_kernel<<<1, 1, 0, stream>>>(acc, (const int*)d_in[24], Bn, (float*)d_out);
}